// MultiHeadAttentionWithRelativePositionalEmbeddings_16896401342847
// MI455X (gfx1250) — compile-verified
//
#include <hip/hip_runtime.h>
#include <math.h>

// Problem constants (match the reference)
#define B_   8
#define N_   512
#define DM   512          // d_model
#define H_   8
#define DK_  64
#define DV_  64
#define HD   512          // H*DK = H*DV
#define ZOFF 511          // NUM_REL // 2

typedef __attribute__((ext_vector_type(2))) float v2f;   // A/B frag for wmma f32 16x16x4
typedef __attribute__((ext_vector_type(8))) float v8f;   // C/D frag (16x16 f32 / 32 lanes)

// D = A(16x4) * B(4x16) + C, full fp32 (CDNA5 V_WMMA_F32_16X16X4_F32)
__device__ __forceinline__ v8f wmma4_f32(v2f a, v2f b, v8f c) {
  return __builtin_amdgcn_wmma_f32_16x16x4_f32(false, a, false, b, (short)0, c,
                                               false, false);
}

// Fragment layouts (ISA 7.12.2, wave32):
//  A (16x4, 2 VGPR): lanes 0-15 -> M=lane,K={0,1}; lanes 16-31 -> M=lane-16,K={2,3}
//  B (4x16, 2 VGPR): VGPR r, lanes 0-15 -> K=r,N=lane; lanes 16-31 -> K=r+2,N=lane-16
//  C/D (16x16, 8 VGPR): VGPR r, lanes 0-15 -> M=r; lanes 16-31 -> M=r+8; N=lane&15

// ---------------------------------------------------------------------------
// Kernel 1: fused QKV projection. Y = X @ W + b, scattered to [b,h,n,d].
// One wave computes a 32(M) x 64(N) tile: per k-step 2 A-frags + 4 B-frags
// feed 8 v_wmma (amortizes VMEM 4x vs 16x16 tiles).
// grid = (HD/64=8, BN/128=32, 3), block = 128 (4 waves, stacked in M)
// ---------------------------------------------------------------------------
__global__ __launch_bounds__(128)
void qkv_proj_kernel(const float* __restrict__ X,
                     const float* __restrict__ Wq, const float* __restrict__ bq,
                     const float* __restrict__ Wk, const float* __restrict__ bk,
                     const float* __restrict__ Wv, const float* __restrict__ bv,
                     float* __restrict__ Q, float* __restrict__ K,
                     float* __restrict__ V)
{
  const int lane = threadIdx.x & 31;
  const int wave = threadIdx.x >> 5;
  const int half = lane >> 4;
  const int l15  = lane & 15;

  const int c0 = blockIdx.x * 64;                 // column tile -> one head
  const int r0 = (blockIdx.y * 4 + wave) * 32;    // row tile (0..4095)

  const float* W; const float* bias; float* Y;
  if (blockIdx.z == 0)      { W = Wq; bias = bq; Y = Q; }
  else if (blockIdx.z == 1) { W = Wk; bias = bk; Y = K; }
  else                      { W = Wv; bias = bv; Y = V; }

  v8f acc[2][4];
#pragma unroll
  for (int m = 0; m < 2; ++m)
#pragma unroll
    for (int n = 0; n < 4; ++n) acc[m][n] = {};

  for (int k = 0; k < DM; k += 4) {
    const int ka = k + half * 2;
    v2f a[2];
#pragma unroll
    for (int m = 0; m < 2; ++m) {
      a[m].x = X[(size_t)(r0 + m * 16 + l15) * DM + ka];
      a[m].y = X[(size_t)(r0 + m * 16 + l15) * DM + ka + 1];
    }
    v2f bfr[4];
#pragma unroll
    for (int n = 0; n < 4; ++n) {
      bfr[n].x = W[(size_t)(ka + 0) * HD + c0 + n * 16 + l15];
      bfr[n].y = W[(size_t)(ka + 1) * HD + c0 + n * 16 + l15];
    }
#pragma unroll
    for (int m = 0; m < 2; ++m)
#pragma unroll
      for (int n = 0; n < 4; ++n) acc[m][n] = wmma4_f32(a[m], bfr[n], acc[m][n]);
  }

  const int h = blockIdx.x;                       // 64-wide tile == one head
#pragma unroll
  for (int n = 0; n < 4; ++n) {
    const int d  = n * 16 + l15;
    const float bb = bias[c0 + d];
#pragma unroll
    for (int m = 0; m < 2; ++m)
#pragma unroll
      for (int r = 0; r < 8; ++r) {
        const int row = r0 + m * 16 + r + half * 8;   // global row in [0, B*N)
        const int b = row >> 9, i = row & 511;
        Y[((size_t)((b * H_ + h) * N_) + i) * DK_ + d] = acc[m][n][r] + bb;
      }
  }
}

// ---------------------------------------------------------------------------
// Kernel 2: per (b, h, 16-row i-tile) attention.
//  Q tile staged Global->LDS with GLOBAL_LOAD_ASYNC_TO_LDS_B128 (ASYNCcnt)
//  scores(16x512) = Q_tile @ K^T (WMMA, 16x64 j-chunks) + P_k[rel]*scale -> LDS
//  softmax rows in LDS
//  out = probs @ V (WMMA) + sum_j probs[i,j]*P_v[511+j-i] (VALU, Toeplitz)
// grid = (N/16=32, H=8, B=8), block = 128
// ---------------------------------------------------------------------------
#define PROBS_LD (N_ + 4)   // pad LDS rows: stride 516 floats -> conflict-free

__global__ __launch_bounds__(128)
void attn_kernel(const float* __restrict__ Q, const float* __restrict__ K,
                 const float* __restrict__ V, const float* __restrict__ Pk,
                 const float* __restrict__ Pv, float* __restrict__ OH)
{
  __shared__ __align__(16) float qtile[16 * DK_];   // 4 KB
  __shared__ float probs[16][PROBS_LD];             // ~33 KB

  const int lane = threadIdx.x & 31;
  const int wave = threadIdx.x >> 5;
  const int half = lane >> 4;
  const int l15  = lane & 15;

  const int i0 = blockIdx.x * 16;
  const int h  = blockIdx.y;
  const int b  = blockIdx.z;
  const float scale = 0.125f;                       // 1/sqrt(DK)

  const float* Qb = Q + (size_t)((b * H_ + h) * N_) * DK_;
  const float* Kb = K + (size_t)((b * H_ + h) * N_) * DK_;
  const float* Vb = V + (size_t)((b * H_ + h) * N_) * DV_;

  // ---- async-stage the 16x64 Q tile (contiguous 4 KB) into LDS ----
  {
    const unsigned long long gbase =
        (unsigned long long)(const void*)(Qb + (size_t)i0 * DK_);
#pragma unroll
    for (int it = 0; it < 2; ++it) {
      const int fo = (threadIdx.x + it * 128) * 4;         // float index, x4 lanes
      const unsigned lds  = (unsigned)(unsigned long long)(const void*)&qtile[fo];
      const unsigned long long ga = gbase + (unsigned long long)fo * 4ull;
      asm volatile("global_load_async_to_lds_b128 %0, %1, off"
                   :: "v"(lds), "v"(ga) : "memory");
    }
    asm volatile("s_wait_asynccnt 0x0" ::: "memory");
  }
  __syncthreads();

  // ---- scores: each wave covers 64-wide j-chunks {wave, wave+4} ----
  for (int jc = wave; jc < N_ / 64; jc += 4) {
    const int j0 = jc * 64;
    v8f acc[4];
#pragma unroll
    for (int n = 0; n < 4; ++n) acc[n] = {};

    for (int k = 0; k < DK_; k += 4) {
      const int ka = k + half * 2;
      v2f a;  a.x = qtile[l15 * DK_ + ka];
              a.y = qtile[l15 * DK_ + ka + 1];
      v2f bf[4];
#pragma unroll
      for (int n = 0; n < 4; ++n) {                 // B[kk,j] = K[j,kk]
        bf[n].x = Kb[(size_t)(j0 + n * 16 + l15) * DK_ + ka];
        bf[n].y = Kb[(size_t)(j0 + n * 16 + l15) * DK_ + ka + 1];
      }
#pragma unroll
      for (int n = 0; n < 4; ++n) acc[n] = wmma4_f32(a, bf[n], acc[n]);
    }
#pragma unroll
    for (int n = 0; n < 4; ++n) {
      const int j = j0 + n * 16 + l15;
#pragma unroll
      for (int r = 0; r < 8; ++r) {
        const int il  = r + half * 8;
        const int rel = ZOFF + j - (i0 + il);       // in [0, 1022]
        probs[il][j] = (acc[n][r] + Pk[rel * H_ + h]) * scale;
      }
    }
  }
  __syncthreads();

  // ---- softmax: 8 threads per row (sub-groups live inside one wave32) ----
  {
    const int row = threadIdx.x >> 3;
    const int sub = threadIdx.x & 7;
    float m = -3.0e38f;
    for (int j = sub; j < N_; j += 8) m = fmaxf(m, probs[row][j]);
#pragma unroll
    for (int off = 4; off >= 1; off >>= 1) m = fmaxf(m, __shfl_xor(m, off, 8));
    float s = 0.f;
    for (int j = sub; j < N_; j += 8) {
      float e = __expf(probs[row][j] - m);
      probs[row][j] = e;
      s += e;
    }
#pragma unroll
    for (int off = 4; off >= 1; off >>= 1) s += __shfl_xor(s, off, 8);
    const float inv = 1.f / s;
    for (int j = sub; j < N_; j += 8) probs[row][j] *= inv;
  }
  __syncthreads();

  // ---- probs @ V : each wave owns a 16-wide d slice ----
  const int d0 = wave * 16;
  v8f acc = {};
  for (int k = 0; k < N_; k += 4) {
    const int ka = k + half * 2;
    v2f a;  a.x = probs[l15][ka];
            a.y = probs[l15][ka + 1];
    v2f bf; bf.x = Vb[(size_t)(ka + 0) * DV_ + d0 + l15];
            bf.y = Vb[(size_t)(ka + 1) * DV_ + d0 + l15];
    acc = wmma4_f32(a, bf, acc);
  }

  // ---- relative-value (Toeplitz) term on VALU + store ----
  const int d = d0 + l15;
#pragma unroll
  for (int r = 0; r < 8; ++r) {
    const int il    = r + half * 8;
    const int i_abs = i0 + il;
    const float* Pvp = Pv + (size_t)(ZOFF - i_abs) * HD + h * DV_ + d;
    float sum = 0.f;
    for (int j = 0; j < N_; ++j)
      sum += probs[il][j] * Pvp[(size_t)j * HD];
    OH[(size_t)(b * N_ + i_abs) * HD + h * DV_ + d] = acc[r] + sum;
  }
}

// ---------------------------------------------------------------------------
// Kernel 3: output projection. out = OH @ Wo + bo (row-major [4096,512]).
// Same 32x64-per-wave tiling as kernel 1.
// grid = (DM/64=8, 32, 1), block = 128
// ---------------------------------------------------------------------------
__global__ __launch_bounds__(128)
void out_proj_kernel(const float* __restrict__ OH, const float* __restrict__ Wo,
                     const float* __restrict__ bo, float* __restrict__ Out)
{
  const int lane = threadIdx.x & 31;
  const int wave = threadIdx.x >> 5;
  const int half = lane >> 4;
  const int l15  = lane & 15;

  const int c0 = blockIdx.x * 64;
  const int r0 = (blockIdx.y * 4 + wave) * 32;

  v8f acc[2][4];
#pragma unroll
  for (int m = 0; m < 2; ++m)
#pragma unroll
    for (int n = 0; n < 4; ++n) acc[m][n] = {};

  for (int k = 0; k < HD; k += 4) {
    const int ka = k + half * 2;
    v2f a[2];
#pragma unroll
    for (int m = 0; m < 2; ++m) {
      a[m].x = OH[(size_t)(r0 + m * 16 + l15) * HD + ka];
      a[m].y = OH[(size_t)(r0 + m * 16 + l15) * HD + ka + 1];
    }
    v2f bf[4];
#pragma unroll
    for (int n = 0; n < 4; ++n) {
      bf[n].x = Wo[(size_t)(ka + 0) * DM + c0 + n * 16 + l15];
      bf[n].y = Wo[(size_t)(ka + 1) * DM + c0 + n * 16 + l15];
    }
#pragma unroll
    for (int m = 0; m < 2; ++m)
#pragma unroll
      for (int n = 0; n < 4; ++n) acc[m][n] = wmma4_f32(a[m], bf[n], acc[m][n]);
  }

#pragma unroll
  for (int n = 0; n < 4; ++n) {
    const int col = c0 + n * 16 + l15;
    const float bb = bo[col];
#pragma unroll
    for (int m = 0; m < 2; ++m)
#pragma unroll
      for (int r = 0; r < 8; ++r) {
        const int row = r0 + m * 16 + r + half * 8;
        Out[(size_t)row * DM + col] = acc[m][n][r] + bb;
      }
  }
}

// ---------------------------------------------------------------------------
extern "C" void kernel_launch(void* const* d_in, const int* in_sizes, int n_in,
                              void* d_out, int out_size, void* d_ws, size_t ws_size,
                              hipStream_t stream) {
  const float* X  = (const float*)d_in[0];
  const float* Wq = (const float*)d_in[1];
  const float* bq = (const float*)d_in[2];
  const float* Wk = (const float*)d_in[3];
  const float* bk = (const float*)d_in[4];
  const float* Wv = (const float*)d_in[5];
  const float* bv = (const float*)d_in[6];
  const float* Wo = (const float*)d_in[7];
  const float* bo = (const float*)d_in[8];
  const float* Pk = (const float*)d_in[9];
  const float* Pv = (const float*)d_in[10];
  float* out = (float*)d_out;

  const size_t qkv_elems = (size_t)B_ * H_ * N_ * DK_;   // 2,097,152 floats
  float* Q  = (float*)d_ws;
  float* K  = Q + qkv_elems;
  float* V  = K + qkv_elems;
  float* OH = V + qkv_elems;                              // [B, N, H*DV]

  // 1) QKV projections (z selects which weight / destination)
  qkv_proj_kernel<<<dim3(HD / 64, (B_ * N_) / 128, 3), 128, 0, stream>>>(
      X, Wq, bq, Wk, bk, Wv, bv, Q, K, V);

  // 2) attention with relative score bias + relative value mixing
  attn_kernel<<<dim3(N_ / 16, H_, B_), 128, 0, stream>>>(Q, K, V, Pk, Pv, OH);

  // 3) output projection
  out_proj_kernel<<<dim3(DM / 64, (B_ * N_) / 128, 1), 128, 0, stream>>>(
      OH, Wo, bo, out);
}